// CubePadding_69243462746691
// MI455X (gfx1250) — compile-verified
//
#include <hip/hip_runtime.h>
#include <hip/hip_bf16.h>

// Problem constants (match reference)
#define PP 2
#define BB 4
#define CC 64
#define HH 128
#define WW 128
#define OH (HH + 2 * PP)   // 132
#define OW (WW + 2 * PP)   // 132
#define B6C (BB * 6 * CC)  // 1536
#define FACE_IN (HH * WW)      // 16384
#define FACE_OUT (OH * OW)     // 17424
#define HALO (2 * PP * OW + 2 * PP * HH)  // 1040 halo elements per (b,face,c)

// -------------------------------------------------------------------------
// Interior copy: rows [P, P+H) x cols [P, P+W) of every output face are an
// identity copy of the input face. Moved via the CDNA5 async LDS engines:
//   global -> LDS : global_load_async_to_lds_b128 (512 B / wave / row)
//   LDS -> global : global_store_async_from_lds_b64 (out is only 8B aligned)
// One wave handles 4 rows (2 KB LDS), one s_wait_asynccnt between phases.
// Total rows = B*6*C*H = 196608; 8 waves/block * 4 rows = 32 rows/block
// -> 6144 blocks exactly, no bounds checks.
// -------------------------------------------------------------------------
__global__ __launch_bounds__(256) void cubepad_interior_async(
    const float* __restrict__ x, float* __restrict__ out) {
  __shared__ __align__(16) unsigned char smem[8 * 2048];

  const int lane = threadIdx.x & 31;
  const int wave = threadIdx.x >> 5;
  const int row0 = (blockIdx.x * 8 + wave) * 4;  // first input row for wave

  const unsigned int ldsBase =
      (unsigned int)(size_t)(&smem[wave * 2048]);

  // Phase 1: issue 4 async row loads (each: 32 lanes x 16 B = 512 B row)
  #pragma unroll
  for (int j = 0; j < 4; ++j) {
    const int r = row0 + j;
    const float* src = x + (size_t)r * WW + lane * 4;
    const unsigned int lds = ldsBase + j * 512 + lane * 16;
    asm volatile("global_load_async_to_lds_b128 %0, %1, off"
                 :: "v"(lds), "v"((unsigned long long)(size_t)src)
                 : "memory");
  }

  // Wait for all async loads to land in LDS.
  asm volatile("s_wait_asynccnt 0" ::: "memory");

  // Phase 2: issue 8 async stores (2 x b64 per row; 8-byte aligned dest).
  #pragma unroll
  for (int j = 0; j < 4; ++j) {
    const int r = row0 + j;
    const int bfc = r >> 7;        // r / H  (H = 128)
    const int h = r & (HH - 1);    // r % H
    float* dstRow = out + (size_t)bfc * FACE_OUT + (size_t)(h + PP) * OW + PP;
    #pragma unroll
    for (int k = 0; k < 2; ++k) {
      float* dst = dstRow + k * 64 + lane * 2;
      const unsigned int lds = ldsBase + j * 512 + k * 256 + lane * 8;
      asm volatile("global_store_async_from_lds_b64 %0, %1, off"
                   :: "v"((unsigned long long)(size_t)dst), "v"(lds)
                   : "memory");
    }
  }

  // Ensure LDS stays valid until async stores complete.
  asm volatile("s_wait_asynccnt 0" ::: "memory");
}

// -------------------------------------------------------------------------
// Halo gather: maps each padded border element (top/bottom strips, left/right
// strips, replicated corners) to its source face element.
// Face order (dim 1 of x): 0=back 1=down 2=front 3=left 4=right 5=top.
// -------------------------------------------------------------------------
__device__ __forceinline__ void cubepad_src_map(int f, int r, int cc,
                                                int& sf, int& sr, int& sc) {
  if (r < PP) {  // top strip (corners replicate edge column via clamp)
    const int p = r;
    const int w = min(max(cc - PP, 0), WW - 1);
    switch (f) {
      case 0:  sf = 5; sr = PP - 1 - p;  sc = w;           break;
      case 1:  sf = 2; sr = HH - PP + p; sc = w;           break;
      case 2:  sf = 5; sr = HH - PP + p; sc = w;           break;
      case 3:  sf = 5; sr = w;           sc = p;           break;
      case 4:  sf = 5; sr = w;           sc = WW - 1 - p;  break;
      default: sf = 0; sr = PP - 1 - p;  sc = w;           break;
    }
  } else if (r >= PP + HH) {  // bottom strip
    const int p = r - PP - HH;
    const int w = min(max(cc - PP, 0), WW - 1);
    switch (f) {
      case 0:  sf = 1; sr = HH - 1 - p; sc = w;            break;
      case 1:  sf = 0; sr = HH - 1 - p; sc = w;            break;
      case 2:  sf = 1; sr = p;          sc = w;            break;
      case 3:  sf = 1; sr = w;          sc = PP - 1 - p;   break;
      case 4:  sf = 1; sr = w;          sc = WW - PP + p;  break;
      default: sf = 2; sr = p;          sc = w;            break;
    }
  } else if (cc < PP) {  // left strip
    const int h = r - PP, q = cc;
    switch (f) {
      case 0:  sf = 4; sr = h;          sc = WW - PP + q;  break;
      case 1:  sf = 3; sr = HH - 1 - q; sc = h;            break;
      case 2:  sf = 3; sr = h;          sc = WW - PP + q;  break;
      case 3:  sf = 0; sr = h;          sc = WW - PP + q;  break;
      case 4:  sf = 2; sr = h;          sc = WW - PP + q;  break;
      default: sf = 3; sr = q;          sc = h;            break;
    }
  } else if (cc >= PP + WW) {  // right strip
    const int h = r - PP, q = cc - PP - WW;
    switch (f) {
      case 0:  sf = 3; sr = h;           sc = q;  break;
      case 1:  sf = 4; sr = HH - PP + q; sc = h;  break;
      case 2:  sf = 4; sr = h;           sc = q;  break;
      case 3:  sf = 2; sr = h;           sc = q;  break;
      case 4:  sf = 0; sr = h;           sc = q;  break;
      default: sf = 4; sr = PP - 1 - q;  sc = h;  break;
    }
  } else {  // interior (not used by halo kernel, kept for completeness)
    sf = f; sr = r - PP; sc = cc - PP;
  }
}

__global__ __launch_bounds__(256) void cubepad_halo(
    const float* __restrict__ x, float* __restrict__ out) {
  const int tid = blockIdx.x * blockDim.x + threadIdx.x;
  if (tid >= B6C * HALO) return;

  const int bfc = tid / HALO;
  const int t = tid - bfc * HALO;

  int r, cc;
  if (t < 2 * PP * OW) {                 // top+bottom full-width rows
    const int rowSel = t / OW;           // 0,1 -> rows 0,1 ; 2,3 -> 130,131
    cc = t - rowSel * OW;
    r = (rowSel < PP) ? rowSel : (HH + rowSel);
  } else {                               // left/right columns of middle rows
    const int t2 = t - 2 * PP * OW;
    const int h = t2 >> 2;               // t2 / (2P)
    const int k = t2 & 3;                // t2 % (2P)
    cc = (k < PP) ? k : (PP + WW + (k - PP));
    r = h + PP;
  }

  const int b = bfc / (6 * CC);
  const int rem = bfc - b * (6 * CC);
  const int f = rem / CC;
  const int c = rem - f * CC;

  int sf, sr, sc;
  cubepad_src_map(f, r, cc, sf, sr, sc);

  const int srcIdx = ((b * 6 + sf) * CC + c) * FACE_IN + sr * WW + sc;
  const int dstIdx = bfc * FACE_OUT + r * OW + cc;
  out[dstIdx] = x[srcIdx];
}

extern "C" void kernel_launch(void* const* d_in, const int* in_sizes, int n_in,
                              void* d_out, int out_size, void* d_ws, size_t ws_size,
                              hipStream_t stream) {
  (void)in_sizes; (void)n_in; (void)out_size; (void)d_ws; (void)ws_size;
  const float* x = (const float*)d_in[0];
  float* out = (float*)d_out;

  // Interior: 196608 rows / (8 waves * 4 rows) = 6144 blocks exactly.
  cubepad_interior_async<<<6144, 256, 0, stream>>>(x, out);

  // Halo: 1536 * 1040 = 1,597,440 elements -> 6240 blocks of 256 exactly.
  cubepad_halo<<<6240, 256, 0, stream>>>(x, out);
}